// DistMul_17815524343862
// MI455X (gfx1250) — compile-verified
//
#include <hip/hip_runtime.h>
#include <hip/hip_bf16.h>
#include <stdint.h>

// DistMult edge scoring: out[e] = sigmoid( sum_d h[u[e],d] * R[etype[e],d] * h[v[e],d] )
// Gather-bound (h fits in 192MB L2). One wave32 per edge, coalesced b128 row loads,
// rel table staged to LDS via the CDNA5 Tensor Data Mover, global_prefetch for latency.

#define D_DIM    384
#define N_ETYPES 8
#define BLOCK    256      // 8 wave32 per block
#define GRID     1024     // 8192 waves total, grid-stride over 250k edges

typedef unsigned int u32x4 __attribute__((ext_vector_type(4)));
typedef int          i32x4 __attribute__((ext_vector_type(4)));
typedef int          i32x8 __attribute__((ext_vector_type(8)));

#define AS1V __attribute__((address_space(1))) void
#define AS3V __attribute__((address_space(3))) void

__global__ __launch_bounds__(BLOCK) void distmult_score_kernel(
    const float* __restrict__ h,
    const int*   __restrict__ u,
    const int*   __restrict__ v,
    const int*   __restrict__ et,
    const float* __restrict__ rel,
    float*       __restrict__ out,
    int nEdges)
{
  __shared__ float srel[N_ETYPES * D_DIM];   // 12 KB relation table

  const int tid  = threadIdx.x;
  const int lane = tid & 31;

  // ---- Stage rel_weight (8 x 384 f32 = 12KB) into LDS ----
#if __has_builtin(__builtin_amdgcn_tensor_load_to_lds) && __has_builtin(__builtin_amdgcn_s_wait_tensorcnt)
  if (tid < 32) {   // one wave issues the TDM descriptor for the whole workgroup
    unsigned lds_addr = (unsigned)(uintptr_t)(AS3V*)&srel[0];
    unsigned long long ga = (unsigned long long)(uintptr_t)rel;
    // D# group0: count=1 | lds_addr | global_addr[56:0] | type=2 ("image")
    u32x4 g0 = {0u, 0u, 0u, 0u};
    g0[0] = 1u;
    g0[1] = lds_addr;
    g0[2] = (unsigned)(ga & 0xFFFFFFFFull);
    g0[3] = (unsigned)((ga >> 32) & 0x1FFFFFFull) | (2u << 30);
    // D# group1: data_size=4B, tensor_dim0=384, tensor_dim1=8, tile 384x8, stride0=384
    i32x8 g1 = {0, 0, 0, 0, 0, 0, 0, 0};
    g1[0] = (2 << 16);                        // data_size code 2 = 4 bytes
    g1[1] = (D_DIM & 0xFFFF) << 16;           // tensor_dim0[15:0] @ bits 63:48
    g1[2] = (N_ETYPES & 0xFFFF) << 16;        // tensor_dim0 hi=0 | tensor_dim1[15:0]
    g1[3] = (D_DIM << 16);                    // tensor_dim1 hi=0 | tile_dim0=384
    g1[4] = N_ETYPES;                         // tile_dim1=8, tile_dim2=0
    g1[5] = D_DIM;                            // tensor_dim0_stride[31:0]
    i32x4 z4 = {0, 0, 0, 0};
#if __clang_major__ >= 23
    i32x8 z8 = {0, 0, 0, 0, 0, 0, 0, 0};
    __builtin_amdgcn_tensor_load_to_lds(g0, g1, z4, z4, z8, 0);
#else
    __builtin_amdgcn_tensor_load_to_lds(g0, g1, z4, z4, 0);
#endif
    __builtin_amdgcn_s_wait_tensorcnt(0);
  }
#elif __has_builtin(__builtin_amdgcn_global_load_async_to_lds_b128)
  for (int i = tid; i < (N_ETYPES * D_DIM) / 4; i += BLOCK) {
    __builtin_amdgcn_global_load_async_to_lds_b128(
        (AS1V*)((const float4*)rel + i),
        (AS3V*)((float4*)srel + i), 0, 0);
  }
#if __has_builtin(__builtin_amdgcn_s_wait_asynccnt)
  __builtin_amdgcn_s_wait_asynccnt(0);
#else
  asm volatile("s_wait_asynccnt 0" ::: "memory");
#endif
#else
  for (int i = tid; i < (N_ETYPES * D_DIM) / 4; i += BLOCK)
    ((float4*)srel)[i] = ((const float4*)rel)[i];
#endif
  __syncthreads();

  const float4* __restrict__ h4    = (const float4*)h;
  const float4* __restrict__ srel4 = (const float4*)srel;

  const int gwave  = blockIdx.x * (BLOCK >> 5) + (tid >> 5);
  const int nwaves = gridDim.x * (BLOCK >> 5);

  for (int e = gwave; e < nEdges; e += nwaves) {
    // wave-uniform edge indices -> scalar regs
    const int un = __builtin_amdgcn_readfirstlane(u[e]);
    const int vn = __builtin_amdgcn_readfirstlane(v[e]);
    const int tn = __builtin_amdgcn_readfirstlane(et[e]);
    const float4* pu = h4 + un * (D_DIM / 4);
    const float4* pv = h4 + vn * (D_DIM / 4);

    // Prefetch next edge's rows while we compute this one (per-lane prefetch
    // covers float4 0-31 and 64-95; rows are 1536B = 96 float4).
    const int e2 = e + nwaves;
    if (e2 < nEdges) {
      const float4* pu2 = h4 + u[e2] * (D_DIM / 4);
      const float4* pv2 = h4 + v[e2] * (D_DIM / 4);
      __builtin_prefetch(pu2 + lane, 0, 1);
      __builtin_prefetch(pu2 + lane + 64, 0, 1);
      __builtin_prefetch(pv2 + lane, 0, 1);
      __builtin_prefetch(pv2 + lane + 64, 0, 1);
    }

    // 32 lanes x 3 sweeps x float4 = 384 elements; fully coalesced b128 loads.
    float acc = 0.0f;
#pragma unroll
    for (int it = 0; it < 3; ++it) {
      const int idx = lane + 32 * it;
      float4 a = pu[idx];
      float4 b = pv[idx];
      float4 r = srel4[tn * (D_DIM / 4) + idx];
      acc = fmaf(a.x * r.x, b.x, acc);
      acc = fmaf(a.y * r.y, b.y, acc);
      acc = fmaf(a.z * r.z, b.z, acc);
      acc = fmaf(a.w * r.w, b.w, acc);
    }

    // wave32 butterfly reduction
#pragma unroll
    for (int off = 16; off > 0; off >>= 1)
      acc += __shfl_xor(acc, off, 32);

    if (lane == 0)
      out[e] = 1.0f / (1.0f + __expf(-acc));
  }
}

extern "C" void kernel_launch(void* const* d_in, const int* in_sizes, int n_in,
                              void* d_out, int out_size, void* d_ws, size_t ws_size,
                              hipStream_t stream) {
  const float* h   = (const float*)d_in[0];
  const int*   u   = (const int*)d_in[1];
  const int*   v   = (const int*)d_in[2];
  const int*   et  = (const int*)d_in[3];
  const float* rel = (const float*)d_in[4];
  float*       out = (float*)d_out;
  const int nEdges = in_sizes[1];
  (void)n_in; (void)out_size; (void)d_ws; (void)ws_size;

  distmult_score_kernel<<<dim3(GRID), dim3(BLOCK), 0, stream>>>(
      h, u, v, et, rel, out, nEdges);
}